// TokensChooseMaskedRouter_24223615549932
// MI455X (gfx1250) — compile-verified
//
#include <hip/hip_runtime.h>
#include <hip/hip_bf16.h>
#include <math.h>
#include <stdint.h>

// Problem constants (from reference setup_inputs)
#define GG 2
#define TT 4096
#define HH 2048
#define EE 64
#define CC 128
#define KK 2
#define GT (GG*TT)                 // 8192
#define N_DISPATCH ((size_t)GT*EE*CC)  // 67,108,864

typedef __attribute__((ext_vector_type(2))) float v2f;
typedef __attribute__((ext_vector_type(8))) float v8f;

__device__ __forceinline__ void wait_asynccnt0() {
#if __has_builtin(__builtin_amdgcn_s_wait_asynccnt)
  __builtin_amdgcn_s_wait_asynccnt(0);
#else
  asm volatile("s_wait_asynccnt 0x0" ::: "memory");
#endif
}

// ---------------------------------------------------------------------------
// 0. init small accumulators (z_acc, psum, cnt)
// ---------------------------------------------------------------------------
__global__ void router_init(float* psum, int* cnt, float* zacc) {
  int i = threadIdx.x;
  if (i < GG*EE) { psum[i] = 0.0f; cnt[i] = 0; }
  if (i == 0) zacc[0] = 0.0f;
}

// ---------------------------------------------------------------------------
// 1. zero the entire output (dispatch + combine + 2 scalars), float4 stores
// ---------------------------------------------------------------------------
__global__ void router_zero_out(float* __restrict__ out, size_t n4, size_t n) {
  size_t i = (size_t)blockIdx.x * blockDim.x + threadIdx.x;
  size_t stride = (size_t)gridDim.x * blockDim.x;
  float4 z = make_float4(0.f, 0.f, 0.f, 0.f);
  float4* o4 = (float4*)out;
  for (size_t v = i; v < n4; v += stride) o4[v] = z;
  for (size_t v = n4*4 + i; v < n; v += stride) out[v] = 0.0f;
}

// ---------------------------------------------------------------------------
// 2. GEMM: logits[GT,EE] = X[GT,HH] @ W[HH,EE] + bias  via V_WMMA_F32_16X16X4_F32
//    block = 128 threads (4 waves); each wave owns a 16(M) x 64(N) tile.
//    B staged in LDS in fragment-native paired layout:
//      Bs[kpair][col] = (W[2kp][col], W[2kp+1][col])  -> one ds_load_b64 per frag.
//    bias staged global->LDS with the CDNA5 async-to-LDS path (ASYNCcnt).
// ---------------------------------------------------------------------------
__global__ void router_gemm(const float* __restrict__ X,
                            const float* __restrict__ W,
                            const float* __restrict__ bias,
                            float* __restrict__ logits) {
  __shared__ __align__(16) float2 Bs[32 * EE];  // K-chunk 64 (as 32 pairs) x 64 cols
  __shared__ __align__(16) float  bias_s[EE];
  const int tid  = threadIdx.x;
  const int wave = tid >> 5;
  const int lane = tid & 31;
  const int half = lane >> 4;       // lanes 16..31 hold K+2,K+3 per 16x4 f32 layout
  const int l15  = lane & 15;
  const int koff = half * 2;
  const int rowA = blockIdx.x * 64 + wave * 16 + l15;
  const float* __restrict__ Arow = X + (size_t)rowA * HH;

  // async global->LDS stage of bias: lanes 0..15 copy 16B each (ASYNCcnt path)
  if (tid < 16) {
    const float* g = bias + tid * 4;
    unsigned ldsoff = (unsigned)(uintptr_t)(&bias_s[tid * 4]);
    asm volatile("global_load_async_to_lds_b128 %0, %1, off"
                 :: "v"(ldsoff), "v"(g) : "memory");
  }

  v8f acc[4] = {};   // 4 N-tiles of 16 cols each

  for (int k0 = 0; k0 < HH; k0 += 64) {
    // stage B chunk [k0..k0+63][0..63] into paired LDS layout
    #pragma unroll
    for (int i = tid; i < 32 * EE; i += 128) {
      int kp = i >> 6, c = i & 63;
      const float* w0 = W + (size_t)(k0 + kp * 2) * EE + c;
      Bs[i] = make_float2(w0[0], w0[EE]);
    }
    __syncthreads();
    if (k0 + 64 < HH)   // speculative prefetch of next A chunk
      __builtin_prefetch(Arow + k0 + 64, 0, 0);
    #pragma unroll 4
    for (int kk = 0; kk < 64; kk += 4) {
      float2 av = *(const float2*)(Arow + k0 + kk + koff);
      v2f a; a[0] = av.x; a[1] = av.y;
      const int kp = (kk + koff) >> 1;
      #pragma unroll
      for (int nt = 0; nt < 4; ++nt) {
        float2 bv = Bs[kp * EE + nt * 16 + l15];   // single ds_load_b64
        v2f b; b[0] = bv.x; b[1] = bv.y;
        acc[nt] = __builtin_amdgcn_wmma_f32_16x16x4_f32(
            false, a, false, b, (short)0, acc[nt], false, false);
      }
    }
    __syncthreads();
  }

  // make async bias stage visible to all waves, then barrier
  wait_asynccnt0();
  __syncthreads();

  // D layout: VGPR r, lanes 0-15 -> M=r, N=lane; lanes 16-31 -> M=r+8, N=lane-16
  const int rowD = blockIdx.x * 64 + wave * 16 + half * 8;
  #pragma unroll
  for (int nt = 0; nt < 4; ++nt) {
    int col = nt * 16 + l15;
    float bv = bias_s[col];
    #pragma unroll
    for (int r = 0; r < 8; ++r)
      logits[(size_t)(rowD + r) * EE + col] = acc[nt][r] + bv;
  }
}

// ---------------------------------------------------------------------------
// 3. per-token softmax over E=64, top-2 experts, z-loss accumulation
// ---------------------------------------------------------------------------
__global__ void router_stats(const float* __restrict__ logits,
                             float* __restrict__ probs,
                             float* __restrict__ gate1,
                             int* __restrict__ eidx,
                             float* __restrict__ zacc) {
  int t = blockIdx.x * blockDim.x + threadIdx.x;
  if (t >= GT) return;
  const float* row = logits + (size_t)t * EE;
  float m = -INFINITY;
  for (int e = 0; e < EE; ++e) m = fmaxf(m, row[e]);
  float s = 0.0f;
  for (int e = 0; e < EE; ++e) s += expf(row[e] - m);
  float logz = m + logf(s);
  float invs = 1.0f / s;
  float p1 = -1.0f, p2 = -1.0f; int i1 = 0, i2 = 0;
  for (int e = 0; e < EE; ++e) {
    float p = expf(row[e] - m) * invs;
    probs[(size_t)t * EE + e] = p;
    if (p > p1)      { p2 = p1; i2 = i1; p1 = p; i1 = e; }
    else if (p > p2) { p2 = p;  i2 = e; }
  }
  gate1[t] = p1;
  eidx[2 * t + 0] = i1;
  eidx[2 * t + 1] = i2;
  atomicAdd(zacc, logz * logz);
}

// ---------------------------------------------------------------------------
// 4. per-(g,e) column sum of router_probs (for aux loss)
// ---------------------------------------------------------------------------
__global__ void router_colsum(const float* __restrict__ probs,
                              float* __restrict__ psum) {
  int ge = blockIdx.x;            // 0..127
  int g = ge >> 6, e = ge & 63;
  __shared__ float red[256];
  float s = 0.0f;
  for (int t = threadIdx.x; t < TT; t += 256)
    s += probs[((size_t)g * TT + t) * EE + e];
  red[threadIdx.x] = s;
  __syncthreads();
  for (int off = 128; off > 0; off >>= 1) {
    if (threadIdx.x < off) red[threadIdx.x] += red[threadIdx.x + off];
    __syncthreads();
  }
  if (threadIdx.x == 0) psum[ge] = red[0];
}

// ---------------------------------------------------------------------------
// 5. in-LDS bitonic argsort of 4096 tokens per group, descending gate,
//    ascending token id on ties (matches stable argsort of -gate)
// ---------------------------------------------------------------------------
__global__ void router_sort(const float* __restrict__ gate1,
                            int* __restrict__ perm) {
  int g = blockIdx.x;
  __shared__ float sk[TT];
  __shared__ int   sv[TT];
  for (int i = threadIdx.x; i < TT; i += 1024) { sk[i] = gate1[g * TT + i]; sv[i] = i; }
  __syncthreads();
  for (int k = 2; k <= TT; k <<= 1) {
    for (int j = k >> 1; j > 0; j >>= 1) {
      for (int i = threadIdx.x; i < TT; i += 1024) {
        int l = i ^ j;
        if (l > i) {
          float ka = sk[i], kb = sk[l];
          int   va = sv[i], vb = sv[l];
          bool aFirst = (ka > kb) || (ka == kb && va < vb); // a precedes in final order
          bool dirUp = ((i & k) == 0);
          bool sw = dirUp ? !aFirst : aFirst;
          if (sw) { sk[i] = kb; sk[l] = ka; sv[i] = vb; sv[l] = va; }
        }
      }
      __syncthreads();
    }
  }
  for (int i = threadIdx.x; i < TT; i += 1024) perm[g * TT + i] = sv[i];
}

// ---------------------------------------------------------------------------
// 6. token_priority: one wave per (g,e); wave32 ballot prefix scan over the
//    flattened [K,Tperm] sequence. Also emits selection counts (aux loss).
// ---------------------------------------------------------------------------
__global__ void router_scan(const int* __restrict__ perm,
                            const int* __restrict__ eidx,
                            int* __restrict__ tprio,
                            int* __restrict__ cnt) {
  int ge = blockIdx.x;            // 0..127
  int g = ge >> 6, e = ge & 63;
  int lane = threadIdx.x & 31;
  unsigned lanemask = (1u << lane) - 1u;
  int count = 0;
  for (int base = 0; base < KK * TT; base += 32) {
    int s = base + lane;
    int k = s >> 12;              // slot / T
    int p = s & (TT - 1);
    int tok = perm[g * TT + p];
    int ei = eidx[(g * TT + tok) * 2 + k];
    bool m = (ei == e);
    unsigned mask = (unsigned)__ballot(m);   // wave32: bits [31:0]
    if (m) {
      int pri = count + __popc(mask & lanemask);
      tprio[((size_t)g * TT + tok) * EE + e] = pri;
    }
    count += __popc(mask);
  }
  if (lane == 0) cnt[ge] = count;
}

// ---------------------------------------------------------------------------
// 7. sparse scatter of dispatch_mask / combine_array (16K nonzeros)
// ---------------------------------------------------------------------------
__global__ void router_scatter(const int* __restrict__ eidx,
                               const int* __restrict__ tprio,
                               const float* __restrict__ probs,
                               float* __restrict__ out) {
  int i = blockIdx.x * blockDim.x + threadIdx.x;  // (g*T+t)*2 + k
  if (i >= GT * KK) return;
  int gt = i >> 1, k = i & 1;
  int e = eidx[gt * 2 + k];
  int p = tprio[(size_t)gt * EE + e];
  if (p >= 0 && p < CC) {
    size_t off = ((size_t)gt * EE + e) * CC + p;
    out[off] = 1.0f;                                   // dispatch_mask
    out[N_DISPATCH + off] = probs[(size_t)gt * EE + e]; // combine_array
  }
}

// ---------------------------------------------------------------------------
// 8. finalize scalar losses
// ---------------------------------------------------------------------------
__global__ void router_losses(const float* __restrict__ psum,
                              const int* __restrict__ cnt,
                              const float* __restrict__ zacc,
                              float* __restrict__ out) {
  if (threadIdx.x == 0 && blockIdx.x == 0) {
    float s = 0.0f;
    for (int i = 0; i < GG * EE; ++i)
      s += ((float)cnt[i] / (float)TT) * (psum[i] / (float)TT);
    float aux = (s / (float)(GG * EE)) * (float)EE * (float)EE;
    out[2 * N_DISPATCH + 0] = aux;
    out[2 * N_DISPATCH + 1] = zacc[0] / (float)GT;
  }
}

// ---------------------------------------------------------------------------
extern "C" void kernel_launch(void* const* d_in, const int* in_sizes, int n_in,
                              void* d_out, int out_size, void* d_ws, size_t ws_size,
                              hipStream_t stream) {
  (void)in_sizes; (void)n_in; (void)out_size; (void)ws_size;
  const float* X    = (const float*)d_in[0];   // [G,T,H]
  const float* W    = (const float*)d_in[1];   // [H,E]
  const float* bias = (const float*)d_in[2];   // [E]
  float* out = (float*)d_out;

  // workspace layout
  float* logits = (float*)d_ws;                  // GT*EE
  float* probs  = logits + (size_t)GT * EE;      // GT*EE
  float* gate1  = probs  + (size_t)GT * EE;      // GT
  int*   eidx   = (int*)(gate1 + GT);            // GT*2
  int*   perm   = eidx + (size_t)GT * 2;         // GT
  int*   tprio  = perm + GT;                     // GT*EE
  float* psum   = (float*)(tprio + (size_t)GT * EE); // G*E
  int*   cnt    = (int*)(psum + GG * EE);        // G*E
  float* zacc   = (float*)(cnt + GG * EE);       // 1

  const size_t n_out = 2 * N_DISPATCH + 2;
  const size_t n4 = n_out / 4;

  router_init<<<1, 256, 0, stream>>>(psum, cnt, zacc);
  router_zero_out<<<8192, 256, 0, stream>>>(out, n4, n_out);
  router_gemm<<<GT / 64, 128, 0, stream>>>(X, W, bias, logits);
  router_stats<<<GT / 256, 256, 0, stream>>>(logits, probs, gate1, eidx, zacc);
  router_colsum<<<GG * EE, 256, 0, stream>>>(probs, psum);
  router_sort<<<GG, 1024, 0, stream>>>(gate1, perm);
  router_scan<<<GG * EE, 32, 0, stream>>>(perm, eidx, tprio, cnt);
  router_scatter<<<(GT * KK) / 256, 256, 0, stream>>>(eidx, tprio, probs, out);
  router_losses<<<1, 32, 0, stream>>>(psum, cnt, zacc, out);
}